// Mamba2Block_58256936403466
// MI455X (gfx1250) — compile-verified
//
#include <hip/hip_runtime.h>
#include <math.h>

// ---------------------------------------------------------------------------
// Mamba2 block for MI455X (gfx1250, wave32, WMMA, async-to-LDS staging).
// fp32 end-to-end; GEMMs + chunked SSD scan use V_WMMA_F32_16X16X4_F32.
// ---------------------------------------------------------------------------

typedef __attribute__((ext_vector_type(2))) float v2f;
typedef __attribute__((ext_vector_type(8))) float v8f;
typedef __attribute__((ext_vector_type(4))) int   v4i;

#define B_       2
#define T_       1024
#define DMODEL   2048
#define DINNER   4096
#define DSTATE   64
#define NHEADS_  64
#define HEADDIM_ 64
#define CONVDIM  4224            // DINNER + 2*DSTATE
#define DINPROJ  8384            // 2*DINNER + 2*DSTATE + NHEADS
#define NTOK     (B_ * T_)       // 2048
#define EPS_     1e-5f
#define LSS      68              // scan LDS row stride (even, 16B-aligned, bank-clean)

#if defined(__has_builtin)
# if __has_builtin(__builtin_amdgcn_global_load_async_to_lds_b128)
#  define HAVE_ASYNC_LDS 1
# else
#  define HAVE_ASYNC_LDS 0
# endif
#else
# define HAVE_ASYNC_LDS 0
#endif

#define AS1 __attribute__((address_space(1)))
#define AS3 __attribute__((address_space(3)))

__device__ __forceinline__ float silu_f(float x) { return x / (1.f + __expf(-x)); }

// 16-byte global -> LDS stage (async on CDNA5, sync fallback otherwise)
__device__ __forceinline__ void stage16(const float* g, float* l) {
#if HAVE_ASYNC_LDS
  __builtin_amdgcn_global_load_async_to_lds_b128(
      (AS1 v4i*)(uintptr_t)g,
      (AS3 v4i*)(uint32_t)(uintptr_t)l,
      0, 0);
#else
  *(float4*)l = *(const float4*)g;
#endif
}
__device__ __forceinline__ void stage_wait() {
#if HAVE_ASYNC_LDS
# if __has_builtin(__builtin_amdgcn_s_wait_asynccnt)
  __builtin_amdgcn_s_wait_asynccnt(0);
# else
  asm volatile("s_wait_asynccnt 0" ::: "memory");
# endif
#endif
}

__device__ __forceinline__ v8f wmma4(v2f a, v2f b, v8f c) {
  return __builtin_amdgcn_wmma_f32_16x16x4_f32(false, a, false, b, (short)0, c, false, false);
}

// ---- K=32 micro-GEMM helpers: preload fragments, then WMMA burst ----------

// one shared A tile row-major over K, two B tiles row-major over K
__device__ __forceinline__ void mm32_rr(const float* A, const float* B0, const float* B1,
                                        v8f& c0, v8f& c1) {
  v2f a[8], b0[8], b1[8];
#pragma unroll
  for (int k = 0; k < 8; ++k) {
    a[k]  = *(const v2f*)(A  + 4 * k);
    b0[k] = *(const v2f*)(B0 + 4 * k);
    b1[k] = *(const v2f*)(B1 + 4 * k);
  }
#pragma unroll
  for (int k = 0; k < 8; ++k) { c0 = wmma4(a[k], b0[k], c0); c1 = wmma4(a[k], b1[k], c1); }
}

// A row-major over K; B stored [k][n] with row stride LSS (pointers pre-offset)
__device__ __forceinline__ void mm32_rk(const float* A, const float* B0, const float* B1,
                                        v8f& c0, v8f& c1) {
  v2f a[8], b0[8], b1[8];
#pragma unroll
  for (int k = 0; k < 8; ++k) {
    a[k] = *(const v2f*)(A + 4 * k);
    b0[k].x = B0[(4 * k) * LSS]; b0[k].y = B0[(4 * k + 1) * LSS];
    b1[k].x = B1[(4 * k) * LSS]; b1[k].y = B1[(4 * k + 1) * LSS];
  }
#pragma unroll
  for (int k = 0; k < 8; ++k) { c0 = wmma4(a[k], b0[k], c0); c1 = wmma4(a[k], b1[k], c1); }
}

// A stored [k][m], B stored [k][n], both row stride LSS (pointers pre-offset)
__device__ __forceinline__ void mm32_kk(const float* A, const float* B0, const float* B1,
                                        v8f& c0, v8f& c1) {
  v2f a[8], b0[8], b1[8];
#pragma unroll
  for (int k = 0; k < 8; ++k) {
    a[k].x  = A[(4 * k) * LSS];  a[k].y  = A[(4 * k + 1) * LSS];
    b0[k].x = B0[(4 * k) * LSS]; b0[k].y = B0[(4 * k + 1) * LSS];
    b1[k].x = B1[(4 * k) * LSS]; b1[k].y = B1[(4 * k + 1) * LSS];
  }
#pragma unroll
  for (int k = 0; k < 8; ++k) { c0 = wmma4(a[k], b0[k], c0); c1 = wmma4(a[k], b1[k], c1); }
}

// 2x2 tile micro-GEMM for the big projections (A-frags shared across n-tiles)
__device__ __forceinline__ void mm32_2x2(const float* A0, const float* A1,
                                         const float* B0, const float* B1,
                                         v8f& c00, v8f& c01, v8f& c10, v8f& c11) {
  v2f a0[8], a1[8], b0[8], b1[8];
#pragma unroll
  for (int k = 0; k < 8; ++k) {
    a0[k] = *(const v2f*)(A0 + 4 * k);
    a1[k] = *(const v2f*)(A1 + 4 * k);
    b0[k] = *(const v2f*)(B0 + 4 * k);
    b1[k] = *(const v2f*)(B1 + 4 * k);
  }
#pragma unroll
  for (int k = 0; k < 8; ++k) {
    c00 = wmma4(a0[k], b0[k], c00); c01 = wmma4(a0[k], b1[k], c01);
    c10 = wmma4(a1[k], b0[k], c10); c11 = wmma4(a1[k], b1[k], c11);
  }
}

// ---------------------------------------------------------------------------
// LayerNorm: one block per token row.
// ---------------------------------------------------------------------------
__global__ __launch_bounds__(256) void ln_kernel(const float* __restrict__ x,
                                                 const float* __restrict__ w,
                                                 const float* __restrict__ b,
                                                 float* __restrict__ u) {
  __shared__ float red[256];
  const int row = blockIdx.x, tid = threadIdx.x;
  const float* xr = x + (size_t)row * DMODEL;
  float s = 0.f;
  for (int i = tid; i < DMODEL; i += 256) s += xr[i];
  red[tid] = s; __syncthreads();
  for (int off = 128; off > 0; off >>= 1) { if (tid < off) red[tid] += red[tid + off]; __syncthreads(); }
  const float mu = red[0] / (float)DMODEL;
  __syncthreads();
  float v = 0.f;
  for (int i = tid; i < DMODEL; i += 256) { float d = xr[i] - mu; v += d * d; }
  red[tid] = v; __syncthreads();
  for (int off = 128; off > 0; off >>= 1) { if (tid < off) red[tid] += red[tid + off]; __syncthreads(); }
  const float rstd = rsqrtf(red[0] / (float)DMODEL + EPS_);
  float* ur = u + (size_t)row * DMODEL;
  for (int i = tid; i < DMODEL; i += 256) ur[i] = (xr[i] - mu) * rstd * w[i] + b[i];
}

// ---------------------------------------------------------------------------
// fp32 WMMA GEMM:  C[M,N] = A[M,K] @ W[N,K]^T (+ resid[M,N])
// 128x64 block tile, BK=32, 8 waves, 2x2 16x16 tiles per wave.
// ---------------------------------------------------------------------------
__global__ __launch_bounds__(256) void gemm_wmma_f32(const float* __restrict__ A,
                                                     const float* __restrict__ W,
                                                     const float* __restrict__ resid,
                                                     float* __restrict__ C,
                                                     int M, int N, int K) {
  __shared__ float As[128][36];   // even pad: 8B-aligned v2f frags, 16B rows, bank-clean
  __shared__ float Bs[64][36];
  const int tid = threadIdx.x, lane = tid & 31, wv = tid >> 5;
  const int m0 = blockIdx.y * 128, n0 = blockIdx.x * 64;
  const int wr = wv >> 1, wc = wv & 1;
  const int mt0 = 32 * wr, mt1 = 32 * wr + 16;
  const int nt0 = 32 * wc, nt1 = 32 * wc + 16;
  const int row = lane & 15, koff = (lane >> 4) * 2;

  v8f c00 = {}, c01 = {}, c10 = {}, c11 = {};

  for (int k0 = 0; k0 < K; k0 += 32) {
#pragma unroll
    for (int i = 0; i < 4; ++i) {                    // A: 128x32 floats
      const int idx = tid + i * 256;
      const int r = idx >> 3, c4 = (idx & 7) * 4;
      stage16(A + (size_t)(m0 + r) * K + k0 + c4, &As[r][c4]);
    }
#pragma unroll
    for (int i = 0; i < 2; ++i) {                    // B: 64x32 floats
      const int idx = tid + i * 256;
      const int r = idx >> 3, c4 = (idx & 7) * 4;
      stage16(W + (size_t)(n0 + r) * K + k0 + c4, &Bs[r][c4]);
    }
    if (k0 + 32 < K) {                               // hint next K-slab toward L2
      __builtin_prefetch(A + (size_t)(m0 + (tid >> 1)) * K + k0 + 32, 0, 1);
      __builtin_prefetch(W + (size_t)(n0 + (tid >> 2)) * K + k0 + 32, 0, 1);
    }
    stage_wait();
    __syncthreads();
    mm32_2x2(&As[mt0 + row][koff], &As[mt1 + row][koff],
             &Bs[nt0 + row][koff], &Bs[nt1 + row][koff],
             c00, c01, c10, c11);
    __syncthreads();
  }

  const int ni = lane & 15, mi8 = 8 * (lane >> 4);
  auto store_tile = [&](const v8f& acc, int tm, int tn) {
#pragma unroll
    for (int i = 0; i < 8; ++i) {
      const int m = m0 + tm + mi8 + i, n = n0 + tn + ni;
      float v = acc[i]; if (resid) v += resid[(size_t)m * N + n];
      C[(size_t)m * N + n] = v;
    }
  };
  store_tile(c00, mt0, nt0); store_tile(c01, mt0, nt1);
  store_tile(c10, mt1, nt0); store_tile(c11, mt1, nt1);
}

// ---------------------------------------------------------------------------
// Depthwise causal conv (width 4) + SiLU over the xBC slice of zxbcdt.
// ---------------------------------------------------------------------------
__global__ __launch_bounds__(256) void conv_silu_kernel(const float* __restrict__ zxbcdt,
                                                        const float* __restrict__ cw,
                                                        const float* __restrict__ cb,
                                                        float* __restrict__ xbc) {
  const int idx = blockIdx.x * 256 + threadIdx.x;
  if (idx >= B_ * T_ * CONVDIM) return;
  const int c = idx % CONVDIM;
  const int t = (idx / CONVDIM) % T_;
  const int b = idx / (CONVDIM * T_);
  float s = cb[c];
#pragma unroll
  for (int k = 0; k < 4; ++k) {
    const int ts = t - 3 + k;
    if (ts >= 0)
      s += zxbcdt[(size_t)(b * T_ + ts) * DINPROJ + DINNER + c] * cw[c * 4 + k];
  }
  xbc[(size_t)idx] = silu_f(s);
}

// ---------------------------------------------------------------------------
// dt = softplus(dt_raw + bias); la = dt * A  (A = -exp(A_log)).
// ---------------------------------------------------------------------------
__global__ __launch_bounds__(256) void dt_kernel(const float* __restrict__ zxbcdt,
                                                 const float* __restrict__ dt_bias,
                                                 const float* __restrict__ A_log,
                                                 float* __restrict__ dts,
                                                 float* __restrict__ la) {
  const int idx = blockIdx.x * 256 + threadIdx.x;  // NTOK * NHEADS
  if (idx >= NTOK * NHEADS_) return;
  const int h = idx % NHEADS_;
  const int rowtok = idx / NHEADS_;
  const float v = zxbcdt[(size_t)rowtok * DINPROJ + (DINNER + CONVDIM) + h] + dt_bias[h];
  const float sp = (v > 20.f) ? v : log1pf(__expf(v));
  dts[idx] = sp;
  la[idx] = sp * (-__expf(A_log[h]));
}

// ---------------------------------------------------------------------------
// Chunked SSD scan (chunk L = 64). One block per (b, head); 16 chunks.
// Per chunk: G = C·B^T -> masked decay W; Y = W·X + (e^a C)·H; H <- e^{aL}H + B'^T·X.
// All GEMMs are fp32 WMMA over LDS tiles.
// ---------------------------------------------------------------------------
__global__ __launch_bounds__(256) void ssm_scan_kernel(const float* __restrict__ xbc,
                                                       const float* __restrict__ dtsg,
                                                       const float* __restrict__ lag,
                                                       const float* __restrict__ Dp,
                                                       float* __restrict__ Y) {
  extern __shared__ float sm[];
  float* Xs  = sm;                  // [64][LSS]  x chunk (t x p)
  float* Bsm = Xs + 64 * LSS;       // [64][LSS]  B chunk (t x n)
  float* Csm = Bsm + 64 * LSS;      // [64][LSS]  C chunk (t x n)
  float* Hs  = Csm + 64 * LSS;      // [64][LSS]  state (n x p), persists
  float* Wsm = Hs + 64 * LSS;       // [64][LSS]  decay-masked G (t x s)
  float* aa  = Wsm + 64 * LSS;      // [64] inclusive cumsum of log dA
  float* dts = aa + 64;             // [64]

  const int tid = threadIdx.x, lane = tid & 31, wv = tid >> 5;
  const int b = blockIdx.x >> 6, h = blockIdx.x & 63;
  const float Dh = Dp[h];

  const int tt0 = wv * 2, tt1 = tt0 + 1;      // tm identical for tile pair
  const int tm0 = (tt0 >> 2) * 16;
  const int tn0 = (tt0 & 3) * 16, tn1 = (tt1 & 3) * 16;
  const int row = lane & 15, koff = (lane >> 4) * 2;
  const int ni = lane & 15, mi8 = 8 * (lane >> 4);

  for (int i = tid; i < 64 * LSS; i += 256) Hs[i] = 0.f;

  for (int c = 0; c < 16; ++c) {
    const int tb = c * 64;
    // stage X / B / C chunk tiles (async b128 where available)
#pragma unroll
    for (int i = 0; i < 4; ++i) {
      const int idx = tid + i * 256;           // 0..1023 -> 64 rows x 16 float4
      const int t = idx >> 4, j4 = (idx & 15) * 4;
      const float* base = xbc + (size_t)(b * T_ + tb + t) * CONVDIM;
      stage16(base + h * 64 + j4,          &Xs[t * LSS + j4]);
      stage16(base + DINNER + j4,          &Bsm[t * LSS + j4]);
      stage16(base + DINNER + DSTATE + j4, &Csm[t * LSS + j4]);
    }
    if (tid < 64) {
      const int gi = (b * T_ + tb + tid) * NHEADS_ + h;
      dts[tid] = dtsg[gi];
      aa[tid] = lag[gi];
    }
    stage_wait();
    __syncthreads();
    // inclusive cumsum of aa over 64 entries (Hillis-Steele)
    for (int off = 1; off < 64; off <<= 1) {
      float v = 0.f;
      if (tid < 64 && tid >= off) v = aa[tid - off];
      __syncthreads();
      if (tid < 64) aa[tid] += v;
      __syncthreads();
    }

    // ---- GEMM1: G = C · B^T ; masked decay into Wsm ----
    {
      v8f g0 = {}; v8f g1 = {};
      mm32_rr(&Csm[(tm0 + row) * LSS + koff],
              &Bsm[(tn0 + row) * LSS + koff], &Bsm[(tn1 + row) * LSS + koff], g0, g1);
      mm32_rr(&Csm[(tm0 + row) * LSS + 32 + koff],
              &Bsm[(tn0 + row) * LSS + 32 + koff], &Bsm[(tn1 + row) * LSS + 32 + koff], g0, g1);
#pragma unroll
      for (int i = 0; i < 8; ++i) {
        const int t0 = tm0 + mi8 + i, s0 = tn0 + ni;
        Wsm[t0 * LSS + s0] = (s0 <= t0) ? g0[i] * __expf(aa[t0] - aa[s0]) * dts[s0] : 0.f;
        const int s1 = tn1 + ni;
        Wsm[t0 * LSS + s1] = (s1 <= t0) ? g1[i] * __expf(aa[t0] - aa[s1]) * dts[s1] : 0.f;
      }
    }
    __syncthreads();

    // in-place scale: C_t *= exp(a_t);  B_s *= exp(a63 - a_s) * dt_s
    const float a63 = aa[63];
    for (int i = tid; i < 64 * 64; i += 256) {
      const int t = i >> 6, j = i & 63;
      Csm[t * LSS + j] *= __expf(aa[t]);
      Bsm[t * LSS + j] *= __expf(a63 - aa[t]) * dts[t];
    }
    __syncthreads();

    // ---- GEMM2: Y = W·X + (e^a C)·H ; write + D*x ----
    {
      v8f y0 = {}; v8f y1 = {};
      mm32_rk(&Wsm[(tm0 + row) * LSS + koff],
              &Xs[koff * LSS + tn0 + row], &Xs[koff * LSS + tn1 + row], y0, y1);
      mm32_rk(&Wsm[(tm0 + row) * LSS + 32 + koff],
              &Xs[(32 + koff) * LSS + tn0 + row], &Xs[(32 + koff) * LSS + tn1 + row], y0, y1);
      mm32_rk(&Csm[(tm0 + row) * LSS + koff],
              &Hs[koff * LSS + tn0 + row], &Hs[koff * LSS + tn1 + row], y0, y1);
      mm32_rk(&Csm[(tm0 + row) * LSS + 32 + koff],
              &Hs[(32 + koff) * LSS + tn0 + row], &Hs[(32 + koff) * LSS + tn1 + row], y0, y1);
#pragma unroll
      for (int i = 0; i < 8; ++i) {
        const int t0 = tm0 + mi8 + i;
        const int p0 = tn0 + ni, p1 = tn1 + ni;
        Y[(size_t)(b * T_ + tb + t0) * DINNER + h * 64 + p0] = y0[i] + Dh * Xs[t0 * LSS + p0];
        Y[(size_t)(b * T_ + tb + t0) * DINNER + h * 64 + p1] = y1[i] + Dh * Xs[t0 * LSS + p1];
      }
    }
    __syncthreads();

    // ---- GEMM3: H <- e^{a63} H + B'^T · X ----
    {
      const float eL = __expf(a63);
      v8f h0, h1;
#pragma unroll
      for (int i = 0; i < 8; ++i) {
        h0[i] = eL * Hs[(tm0 + mi8 + i) * LSS + tn0 + ni];
        h1[i] = eL * Hs[(tm0 + mi8 + i) * LSS + tn1 + ni];
      }
      mm32_kk(&Bsm[koff * LSS + tm0 + row],
              &Xs[koff * LSS + tn0 + row], &Xs[koff * LSS + tn1 + row], h0, h1);
      mm32_kk(&Bsm[(32 + koff) * LSS + tm0 + row],
              &Xs[(32 + koff) * LSS + tn0 + row], &Xs[(32 + koff) * LSS + tn1 + row], h0, h1);
      __syncthreads();   // all waves done reading Hs before overwrite
#pragma unroll
      for (int i = 0; i < 8; ++i) {
        Hs[(tm0 + mi8 + i) * LSS + tn0 + ni] = h0[i];
        Hs[(tm0 + mi8 + i) * LSS + tn1 + ni] = h1[i];
      }
    }
    __syncthreads();
  }
}

// ---------------------------------------------------------------------------
// Gated RMSNorm, in place on Y: y = y*silu(z); y *= rsqrt(mean(y^2)+eps)*w
// ---------------------------------------------------------------------------
__global__ __launch_bounds__(256) void gate_rms_kernel(const float* __restrict__ zxbcdt,
                                                       const float* __restrict__ nw,
                                                       float* __restrict__ Y) {
  __shared__ float red[256];
  const int rowtok = blockIdx.x, tid = threadIdx.x;
  const float* zr = zxbcdt + (size_t)rowtok * DINPROJ;
  float* yr = Y + (size_t)rowtok * DINNER;
  float ss = 0.f;
  for (int i = tid; i < DINNER; i += 256) {
    const float v = yr[i] * silu_f(zr[i]);
    yr[i] = v;
    ss += v * v;
  }
  red[tid] = ss; __syncthreads();
  for (int off = 128; off > 0; off >>= 1) { if (tid < off) red[tid] += red[tid + off]; __syncthreads(); }
  const float sc = rsqrtf(red[0] / (float)DINNER + EPS_);
  for (int i = tid; i < DINNER; i += 256) yr[i] = yr[i] * sc * nw[i];
}

// ---------------------------------------------------------------------------
extern "C" void kernel_launch(void* const* d_in, const int* in_sizes, int n_in,
                              void* d_out, int out_size, void* d_ws, size_t ws_size,
                              hipStream_t stream) {
  const float* x          = (const float*)d_in[0];
  const float* ln_w       = (const float*)d_in[1];
  const float* ln_b       = (const float*)d_in[2];
  const float* in_proj_w  = (const float*)d_in[3];
  const float* conv_w     = (const float*)d_in[4];
  const float* conv_b     = (const float*)d_in[5];
  const float* dt_bias    = (const float*)d_in[6];
  const float* A_log      = (const float*)d_in[7];
  const float* Dp         = (const float*)d_in[8];
  const float* norm_w     = (const float*)d_in[9];
  const float* out_proj_w = (const float*)d_in[10];
  float* out = (float*)d_out;

  float* u       = (float*)d_ws;                          // NTOK*DMODEL
  float* zxbcdt  = u + (size_t)NTOK * DMODEL;             // NTOK*DINPROJ
  float* xbc     = zxbcdt + (size_t)NTOK * DINPROJ;       // NTOK*CONVDIM
  float* dts     = xbc + (size_t)NTOK * CONVDIM;          // NTOK*NHEADS
  float* la      = dts + (size_t)NTOK * NHEADS_;          // NTOK*NHEADS
  float* Y       = la + (size_t)NTOK * NHEADS_;           // NTOK*DINNER

  ln_kernel<<<NTOK, 256, 0, stream>>>(x, ln_w, ln_b, u);

  dim3 g1(DINPROJ / 64, NTOK / 128);
  gemm_wmma_f32<<<g1, 256, 0, stream>>>(u, in_proj_w, nullptr, zxbcdt, NTOK, DINPROJ, DMODEL);

  conv_silu_kernel<<<(B_ * T_ * CONVDIM + 255) / 256, 256, 0, stream>>>(zxbcdt, conv_w, conv_b, xbc);
  dt_kernel<<<(NTOK * NHEADS_ + 255) / 256, 256, 0, stream>>>(zxbcdt, dt_bias, A_log, dts, la);

  const size_t smem = (size_t)(5 * 64 * LSS + 128) * sizeof(float);
  ssm_scan_kernel<<<B_ * NHEADS_, 256, smem, stream>>>(xbc, dts, la, Dp, Y);

  gate_rms_kernel<<<NTOK, 256, 0, stream>>>(zxbcdt, norm_w, Y);

  dim3 g2(DMODEL / 64, NTOK / 128);
  gemm_wmma_f32<<<g2, 256, 0, stream>>>(Y, out_proj_w, x, out, NTOK, DMODEL, DINNER);
}